// DuplicationRemovalNetwork_39204461478694
// MI455X (gfx1250) — compile-verified
//
#include <hip/hip_runtime.h>
#include <hip/hip_bf16.h>
#include <math.h>

// N=1024, C=2, D=1024, G=16, DG=64, GEO=64, K=10
#define NB   1024
#define NC   2
#define ND   1024
#define NG   16
#define NDG  64
#define NGEO 64
#define NK   10
#define EPSV 1e-6f

typedef __attribute__((ext_vector_type(2))) float v2f;
typedef __attribute__((ext_vector_type(8))) float v8f;

// ---------------------------------------------------------------------------
// Kernel 1: q/k/pf projections.  One wave (32 threads) computes a 16x16 tile
// of  out = fr @ W^T + b  with V_WMMA_F32_16X16X4_F32, K-loop over 1024.
// fr[c*N+n, d] = f_a[n, c, d].
// blockIdx.z selects which of {WQ, WK, conv_w-as-[1024,1024]} is used.
// ---------------------------------------------------------------------------
__global__ __launch_bounds__(32)
void proj_kernel(const float* __restrict__ f_a,
                 const float* __restrict__ Wq, const float* __restrict__ bq,
                 const float* __restrict__ Wk, const float* __restrict__ bk,
                 const float* __restrict__ Wc, const float* __restrict__ bc,
                 float* __restrict__ outq, float* __restrict__ outk,
                 float* __restrict__ outp)
{
    const int lane = threadIdx.x;
    const int n0 = blockIdx.x * 16;     // output column tile
    const int m0 = blockIdx.y * 16;     // output row tile (cn index)
    const int z  = blockIdx.z;

    const float* W = (z == 0) ? Wq : (z == 1) ? Wk : Wc;
    const float* B = (z == 0) ? bq : (z == 1) ? bk : bc;
    float*       O = (z == 0) ? outq : (z == 1) ? outk : outp;

    const int half = lane >> 4;         // 0: K={0,1}, 1: K={2,3}
    const int l16  = lane & 15;

    // A row for this lane: fr[m0+l16, :]  ->  f_a[box, c, :]
    const int cn  = m0 + l16;
    const int c   = cn >> 10;
    const int box = cn & (NB - 1);
    const float* arow = f_a + ((size_t)box * NC + c) * ND;
    // B[k, n] = W[n, k]; this lane supplies column n = n0+l16
    const float* brow = W + (size_t)(n0 + l16) * ND;

    v8f acc = {};
    for (int k0 = 0; k0 < ND; k0 += 4) {
        v2f a, b;
        a.x = arow[k0 + 2 * half + 0];
        a.y = arow[k0 + 2 * half + 1];
        b.x = brow[k0 + 2 * half + 0];
        b.y = brow[k0 + 2 * half + 1];
        acc = __builtin_amdgcn_wmma_f32_16x16x4_f32(
            false, a, false, b, (short)0, acc, false, false);
    }

    const float bias = B[n0 + l16];
#pragma unroll
    for (int r = 0; r < 8; ++r) {
        const int m = m0 + r + 8 * half;            // C layout: VGPR r -> M=r / r+8
        O[(size_t)m * ND + n0 + l16] = acc[r] + bias;
    }
}

// ---------------------------------------------------------------------------
// Kernel 2: gate.  Streams position_embedding (512 MB) exactly once.
// Each thread handles one (c, m, n): 16 dots of length 64 against WG_w,
// then  log(max(relu(x), eps)) + log_iou  -> wg[(c*16+g), m, n]  (coalesced).
// ---------------------------------------------------------------------------
__global__ __launch_bounds__(256)
void gate_kernel(const float* __restrict__ pe, const float* __restrict__ iou,
                 const float* __restrict__ WGw, const float* __restrict__ WGb,
                 float* __restrict__ wg)
{
    __shared__ float sW[NG * NGEO];
    __shared__ float sB[NG];
    const int tid = threadIdx.x;

    for (int i = tid; i < NG * NGEO; i += 256) sW[i] = WGw[i];
    if (tid < NG) sB[tid] = WGb[tid];
    __syncthreads();

    // blockIdx.x = [c(1b)][m(10b)][ntile(2b)]
    const int ntile = blockIdx.x & 3;
    const int m     = (blockIdx.x >> 2) & (NB - 1);
    const int c     = blockIdx.x >> 12;
    const int n     = ntile * 256 + tid;

    const float* pv = pe + (((size_t)c * NB + m) * NB + n) * NGEO;
    __builtin_prefetch(pv + NGEO * 256, 0, 0);   // hint: next tile of the stream

    float4 p[NGEO / 4];
#pragma unroll
    for (int j = 0; j < NGEO / 4; ++j) p[j] = ((const float4*)pv)[j];

    const float li = (iou[((size_t)c * NB + m) * NB + n] >= EPSV)
                         ? 0.0f : logf(EPSV);

#pragma unroll
    for (int g = 0; g < NG; ++g) {
        float d = sB[g];
        const float* wr = &sW[g * NGEO];
#pragma unroll
        for (int j = 0; j < NGEO / 4; ++j) {
            const float4 w = ((const float4*)wr)[j];
            d += p[j].x * w.x + p[j].y * w.y + p[j].z * w.z + p[j].w * w.w;
        }
        const float rel = fmaxf(d, 0.0f);
        wg[((size_t)(c * NG + g) * NB + m) * NB + n] =
            logf(fmaxf(rel, EPSV)) + li;
    }
}

// ---------------------------------------------------------------------------
// Kernel 3: fused affinity + gate + top-10 + softmax + sparse aggregation.
// Block = (cg, m-tile of 16 rows), 16 waves (512 threads).
//   Phase A : WMMA f32 affinity tile [16 x 1024] * 1/8 into LDS
//   Phase A2: add gate row (coalesced)
//   Phase B : per-wave iterative top-10 (wave32 shfl_xor argmax reductions)
//   Phase C : softmax(10) and gather pf rows -> out[m, c, g*64+o]
// ---------------------------------------------------------------------------
__global__ __launch_bounds__(512)
void attend_kernel(const float* __restrict__ q, const float* __restrict__ kmat,
                   const float* __restrict__ pf, const float* __restrict__ wg,
                   float* __restrict__ out)
{
    __shared__ float s[16][NB];         // 64 KB of the 320 KB WGP LDS
    __shared__ float sq[16][NDG];       // q tile, reused by every wave

    const int tid  = threadIdx.x;
    const int wave = tid >> 5;
    const int lane = tid & 31;
    const int cg = blockIdx.y;          // 0..31
    const int c  = cg >> 4;
    const int g  = cg & 15;
    const int m0 = blockIdx.x * 16;

    for (int i = tid; i < 16 * NDG; i += 512) {
        const int r = i >> 6, col = i & 63;
        sq[r][col] = q[((size_t)c * NB + m0 + r) * ND + g * NDG + col];
    }
    __syncthreads();

    const int half = lane >> 4, l16 = lane & 15;

    // ---- Phase A ----
    for (int t = wave; t < NB / 16; t += 16) {
        const int n0 = t * 16;
        const float* arow = &sq[l16][0];
        const float* brow = kmat + ((size_t)c * NB + n0 + l16) * ND + g * NDG;
        v8f acc = {};
#pragma unroll
        for (int k0 = 0; k0 < NDG; k0 += 4) {
            v2f a, b;
            a.x = arow[k0 + 2 * half + 0];
            a.y = arow[k0 + 2 * half + 1];
            b.x = brow[k0 + 2 * half + 0];
            b.y = brow[k0 + 2 * half + 1];
            acc = __builtin_amdgcn_wmma_f32_16x16x4_f32(
                false, a, false, b, (short)0, acc, false, false);
        }
#pragma unroll
        for (int r = 0; r < 8; ++r)
            s[r + 8 * half][n0 + l16] = acc[r] * 0.125f;   // 1/sqrt(64)
    }
    __syncthreads();

    // ---- Phase A2: add gate+log_iou row (this wave owns row `wave`) ----
    {
        const float* wrow = wg + ((size_t)cg * NB + m0 + wave) * NB;
        for (int j = lane; j < NB; j += 32) s[wave][j] += wrow[j];
    }

    // ---- Phase B: top-10 on row `wave` ----
    float vals[NK];
    int   idxs[NK];
    float* row = s[wave];
    for (int it = 0; it < NK; ++it) {
        float best = -__builtin_inff();
        int   bidx = 0x7fffffff;
        for (int j = lane; j < NB; j += 32) {
            const float v = row[j];
            if (v > best || (v == best && j < bidx)) { best = v; bidx = j; }
        }
#pragma unroll
        for (int off = 16; off > 0; off >>= 1) {
            const float ov = __shfl_xor(best, off, 32);
            const int   oi = __shfl_xor(bidx, off, 32);
            if (ov > best || (ov == best && oi < bidx)) { best = ov; bidx = oi; }
        }
        vals[it] = best;
        idxs[it] = bidx;
        if (lane == 0) row[bidx] = -__builtin_inff();
    }

    // ---- softmax over the 10 selected ----
    const float mx = vals[0];
    float wgt[NK], sum = 0.0f;
#pragma unroll
    for (int i = 0; i < NK; ++i) { wgt[i] = __expf(vals[i] - mx); sum += wgt[i]; }
    const float inv = 1.0f / sum;

    // ---- Phase C: out[m, c, g*64+o] = sum_k sm_k * pf[c*N+n_k, g*64+o] ----
    const int m = m0 + wave;
#pragma unroll
    for (int rep = 0; rep < 2; ++rep) {
        const int o = lane + rep * 32;
        float a = 0.0f;
#pragma unroll
        for (int i = 0; i < NK; ++i)
            a += wgt[i] * pf[((size_t)c * NB + idxs[i]) * ND + g * NDG + o];
        out[((size_t)m * NC + c) * ND + g * NDG + o] = a * inv;
    }
}

// ---------------------------------------------------------------------------
extern "C" void kernel_launch(void* const* d_in, const int* in_sizes, int n_in,
                              void* d_out, int out_size, void* d_ws, size_t ws_size,
                              hipStream_t stream)
{
    const float* f_a = (const float*)d_in[0];
    const float* pe  = (const float*)d_in[1];
    const float* iou = (const float*)d_in[2];
    const float* WGw = (const float*)d_in[3];
    const float* WGb = (const float*)d_in[4];
    const float* WKw = (const float*)d_in[5];
    const float* WKb = (const float*)d_in[6];
    const float* WQw = (const float*)d_in[7];
    const float* WQb = (const float*)d_in[8];
    const float* Cw  = (const float*)d_in[9];   // [G, D/G, D] == [1024, 1024]
    const float* Cb  = (const float*)d_in[10];
    float* out = (float*)d_out;

    float* ws = (float*)d_ws;
    const size_t MAT = (size_t)(NC * NB) * ND;  // 2048*1024
    float* q  = ws;
    float* k  = ws + MAT;
    float* pf = ws + 2 * MAT;
    float* wg = ws + 3 * MAT;                   // [C*G, N, N] = 32M floats

    // 1) q/k/pf projections (fp32 WMMA)
    proj_kernel<<<dim3(ND / 16, (NC * NB) / 16, 3), 32, 0, stream>>>(
        f_a, WQw, WQb, WKw, WKb, Cw, Cb, q, k, pf);

    // 2) gate: stream 512 MB of pe once
    gate_kernel<<<dim3(NC * NB * (NB / 256)), 256, 0, stream>>>(
        pe, iou, WGw, WGb, wg);

    // 3) fused affinity + top-k + softmax + sparse aggregate + grouped conv
    attend_kernel<<<dim3(NB / 16, NC * NG), 512, 0, stream>>>(
        q, k, pf, wg, out);
}